// CrossAttention_69698729279861
// MI455X (gfx1250) — compile-verified
//
#include <hip/hip_runtime.h>

typedef float v2f __attribute__((ext_vector_type(2)));
typedef float v8f __attribute__((ext_vector_type(8)));

#define HW_TOK   784
#define CH       64
#define DQKV     128
#define BATCH    512
#define MT_PER_B 49   // 784 / 16 row-tiles per batch item
#define NTILES   8    // 128 / 16 col-tiles

// out[b, t, d] = sum_c x[b^1, t, c] * W[d, c]   (attn==1 collapses the rest)
__global__ __launch_bounds__(256) void cross_attn_gemm_f32wmma(
    const float* __restrict__ x, const float* __restrict__ W,
    float* __restrict__ out)
{
    const int lane = threadIdx.x & 31;
    const int wave = threadIdx.x >> 5;
    const int wid  = blockIdx.x * 8 + wave;      // global wave id
    const int b    = wid / MT_PER_B;
    const int mt   = wid % MT_PER_B;

    const int m  = lane & 15;    // M (A) / N (B) index within 16-wide tile
    const int kh = lane >> 4;    // K-half selector (lanes 16-31 hold K+2)

    // partner-swap: main/other pairing makes every output row come from x[b^1]
    const float* src = x + (size_t)(b ^ 1) * HW_TOK * CH;

    // A-tile: 16 rows x 64 K, resident in regs. Per ISA 32-bit A layout:
    // lane m holds (row=m, K=kbase+2*kh .. +1) -> one b64 load per K-step.
    const float* arow = src + (size_t)(mt * 16 + m) * CH + kh * 2;
    v2f a[16];
#pragma unroll
    for (int ks = 0; ks < 16; ++ks)
        a[ks] = *(const v2f*)(arow + ks * 4);

#pragma unroll
    for (int nt = 0; nt < NTILES; ++nt) {
        // B = W^T tile: B[k][n] = W[n][k]; lane layout mirrors A, and the
        // two K-values per lane are contiguous along W's C dim -> b64 load.
        const float* brow = W + (size_t)(nt * 16 + m) * CH + kh * 2;
        v8f acc = {};
#pragma unroll
        for (int ks = 0; ks < 16; ++ks) {
            v2f bf = *(const v2f*)(brow + ks * 4);
            acc = __builtin_amdgcn_wmma_f32_16x16x4_f32(
                /*neg_a=*/false, a[ks], /*neg_b=*/false, bf,
                /*c_mod=*/(short)0, acc, /*reuse_a=*/false, /*reuse_b=*/false);
        }
        // C/D layout: VGPR r -> row (mt*16 + kh*8 + r), col (nt*16 + m)
        float* orow = out + (size_t)b * HW_TOK * DQKV
                          + (size_t)(mt * 16 + kh * 8) * DQKV
                          + nt * 16 + m;
#pragma unroll
        for (int r = 0; r < 8; ++r)
            orow[(size_t)r * DQKV] = acc[r];
    }
}

__global__ void fill_ones(float* __restrict__ p, int n)
{
    int i = blockIdx.x * blockDim.x + threadIdx.x;
    if (i < n) p[i] = 1.0f;
}

extern "C" void kernel_launch(void* const* d_in, const int* in_sizes, int n_in,
                              void* d_out, int out_size, void* d_ws, size_t ws_size,
                              hipStream_t stream)
{
    const float* x = (const float*)d_in[0];   // [512, 784, 64] f32
    const float* W = (const float*)d_in[1];   // [128, 64] f32
    float* out = (float*)d_out;

    const int total_waves = BATCH * MT_PER_B;         // 25088
    const int blocks = total_waves / 8;               // 3136 full blocks, EXEC all-1s
    cross_attn_gemm_f32wmma<<<blocks, 256, 0, stream>>>(x, W, out);

    float* attn = out + (size_t)BATCH * HW_TOK * DQKV;
    const int n_attn = BATCH * HW_TOK;                // 401408
    fill_ones<<<(n_attn + 255) / 256, 256, 0, stream>>>(attn, n_attn);
}